// RetinaNet_81269371175166
// MI455X (gfx1250) — compile-verified
//
#include <hip/hip_runtime.h>
#include <hip/hip_bf16.h>

// ---------------------------------------------------------------------------
// RetinaNet loss (class + bbox) for MI455X / gfx1250, wave32.
//
// Bandwidth-bound: ~140 MB of traffic (keys-cached path) -> ~6 us at
// 23.3 TB/s. Compute is free, so everything is fused into 4 streaming/tiny
// kernels. Wave reductions use V_WMMA_F32_16X16X4_F32 (A = partials,
// B = ones) so the reduction rides the matrix pipe and co-executes with the
// VMEM stream (XDL ops are tracked like TRANS ops and dual-issue with VALU).
//
// Pass A streams matches+logits once: log-sum-exp, positive CE (WMMA wave
//   reduce -> 1 atomic/block), per-block positive counts, LDS 4096-bin
//   histogram of the order-preserving uint key of logp0; optionally caches
//   the key per anchor (8 MB) so pass C never touches logits again.
// Pass B (1 block) scans block counts and finds the histogram threshold.
// Pass C streams matches+keys: rank-ordered smooth-L1 for positives,
//   threshold-selection of the 40960 hardest negatives into a pool.
// Pass D subsamples 2048 pool entries and combines the scalars.
//
// Approximations (not bit-reproducible vs JAX, by necessity):
//  * top_k(40960) over bg-prob via radix-histogram threshold selection
//    (ties in the threshold bin resolved by atomic quota).
//  * jax.random.permutation(key 42) replaced by the affine bijection
//    j = (48271*k + 7) mod 40960 (48271 odd, coprime to 5 -> bijection).
// ---------------------------------------------------------------------------

#define NCLS   8
#define NPOS   2048
#define POOLSZ (2048 * 20)   // 40960
#define NBINS  4096
#define BLK    256

typedef __attribute__((ext_vector_type(2))) float v2f;
typedef __attribute__((ext_vector_type(8))) float v8f;

// Wave32 sum-reduction on the matrix pipe.
// A (16x4 f32): vgpr0 = per-lane x  -> A[m][0]=x[m], A[m][2]=x[16+m]; vgpr1=0.
// B = ones  =>  D[m][n] = x[m] + x[16+m]  (row sums, replicated over n).
// Summing the 8 D vgprs per lane gives half-sums; one shfl_xor(16) finishes.
// EXEC must be all ones at the call site (uniform control flow only).
__device__ __forceinline__ float wave_reduce_wmma(float x) {
  v2f a; a[0] = x;    a[1] = 0.0f;
  v2f b; b[0] = 1.0f; b[1] = 1.0f;
  v8f c = {};
  c = __builtin_amdgcn_wmma_f32_16x16x4_f32(false, a, false, b,
                                            (short)0, c, false, false);
  float acc = c[0] + c[1] + c[2] + c[3] + c[4] + c[5] + c[6] + c[7];
  acc += __shfl_xor(acc, 16, 32);
  return acc;   // total in every lane
}

__device__ __forceinline__ unsigned order_key(float f) {
  unsigned u = __float_as_uint(f);
  return (u & 0x80000000u) ? ~u : (u | 0x80000000u);  // monotone uint key
}

__device__ __forceinline__ float key_to_float(unsigned k) {
  unsigned u = (k & 0x80000000u) ? (k ^ 0x80000000u) : ~k;  // inverse map
  return __uint_as_float(u);
}

__global__ void __launch_bounds__(BLK) zero_ws(unsigned* w, int n) {
  int i = blockIdx.x * BLK + threadIdx.x;
  if (i < n) w[i] = 0u;
}

// Pass A: stream matches+logits once.
__global__ void __launch_bounds__(BLK) pass_a(const int* __restrict__ matches,
                                              const float* __restrict__ logits,
                                              unsigned* __restrict__ hist,
                                              unsigned* __restrict__ blockcounts,
                                              float* __restrict__ pos_sum,
                                              unsigned* __restrict__ keys, // may be null
                                              int n) {
  __shared__ unsigned hlds[NBINS];
  __shared__ float    wsum[BLK / 32];
  __shared__ unsigned wpos[BLK / 32];
  for (int b = threadIdx.x; b < NBINS; b += BLK) hlds[b] = 0u;
  __syncthreads();

  int i = blockIdx.x * BLK + threadIdx.x;
  int m = 0;
  float ce = 0.0f;
  unsigned key = 0xFFFFFFFFu;   // sentinel: not a negative anchor
  if (i < n) {
    m = matches[i];
    const float4* row = (const float4*)(logits + (size_t)i * NCLS);
    float4 r0 = row[0], r1 = row[1];
    float l[NCLS] = {r0.x, r0.y, r0.z, r0.w, r1.x, r1.y, r1.z, r1.w};
    float mx = l[0];
#pragma unroll
    for (int k = 1; k < NCLS; ++k) mx = fmaxf(mx, l[k]);
    float s = 0.0f;
#pragma unroll
    for (int k = 0; k < NCLS; ++k) s += __expf(l[k] - mx);
    float lse = mx + __logf(s);
    if (m > 0) {
      float lm = l[0];
#pragma unroll
      for (int k = 1; k < NCLS; ++k) lm = (m == k) ? l[k] : lm;
      ce = lse - lm;                       // -logp[target]
    }
    if (m == -1) {
      float logp0 = l[0] - lse;            // monotone with bg_prob
      key = order_key(logp0);
      atomicAdd(&hlds[key >> 20], 1u);
    }
    if (keys) keys[i] = key;               // uniform branch
  }

  float wtot = wave_reduce_wmma(ce);       // uniform flow, EXEC all ones
  unsigned long long bal = __ballot(m > 0);
  int wave = threadIdx.x >> 5, lane = threadIdx.x & 31;
  if (lane == 0) { wsum[wave] = wtot; wpos[wave] = (unsigned)__popcll(bal); }
  __syncthreads();
  if (threadIdx.x == 0) {
    float t = 0.0f; unsigned p = 0u;
    for (int w = 0; w < BLK / 32; ++w) { t += wsum[w]; p += wpos[w]; }
    if (t != 0.0f) atomicAdd(pos_sum, t);
    blockcounts[blockIdx.x] = p;
  }
  __syncthreads();
  for (int b = threadIdx.x; b < NBINS; b += BLK)
    if (hlds[b]) atomicAdd(&hist[b], hlds[b]);
}

// Pass B (1 block): exclusive scan of per-block positive counts (chunked
// Hillis-Steele in LDS), then find the histogram threshold bin + quota.
__global__ void __launch_bounds__(BLK) pass_b(const unsigned* __restrict__ blockcounts,
                                              unsigned* __restrict__ blockoffsets,
                                              int nblocks,
                                              const unsigned* __restrict__ hist,
                                              unsigned* __restrict__ ctrl) {
  __shared__ unsigned tmp[BLK];
  __shared__ unsigned carry;
  if (threadIdx.x == 0) carry = 0u;
  __syncthreads();
  for (int base = 0; base < nblocks; base += BLK) {
    int idx = base + (int)threadIdx.x;
    unsigned v = (idx < nblocks) ? blockcounts[idx] : 0u;
    tmp[threadIdx.x] = v;
    __syncthreads();
    for (int off = 1; off < BLK; off <<= 1) {
      unsigned add = (threadIdx.x >= (unsigned)off) ? tmp[threadIdx.x - off] : 0u;
      __syncthreads();
      tmp[threadIdx.x] += add;
      __syncthreads();
    }
    if (idx < nblocks) blockoffsets[idx] = tmp[threadIdx.x] - v + carry;
    __syncthreads();
    if (threadIdx.x == BLK - 1) carry += tmp[BLK - 1];
    __syncthreads();
  }
  if (threadIdx.x == 0) {
    unsigned cum = 0u, thr = NBINS - 1, quota = 0u;
    for (int b = 0; b < NBINS; ++b) {
      unsigned c = hist[b];
      if (cum + c >= POOLSZ) { thr = (unsigned)b; quota = POOLSZ - cum; break; }
      cum += c;
    }
    ctrl[0] = thr; ctrl[1] = quota;
  }
}

// Pass C: second stream over matches (+cached keys, or logits fallback).
__global__ void __launch_bounds__(BLK) pass_c(const int* __restrict__ matches,
                                              const float* __restrict__ logits,
                                              const unsigned* __restrict__ keys, // may be null
                                              const float* __restrict__ pred_deltas,
                                              const float* __restrict__ target_deltas,
                                              const unsigned* __restrict__ blockoffsets,
                                              const unsigned* __restrict__ ctrl,
                                              unsigned* __restrict__ counters, // [pool, rem]
                                              float* __restrict__ pool_vals,
                                              float* __restrict__ bbox_sum,
                                              int n) {
  __shared__ unsigned woff[BLK / 32];
  int i = blockIdx.x * BLK + threadIdx.x;
  int wave = threadIdx.x >> 5, lane = threadIdx.x & 31;
  int m = (i < n) ? matches[i] : 0;
  bool pos = m > 0;
  unsigned long long bal = __ballot(pos);
  if (lane == 0) woff[wave] = (unsigned)__popcll(bal);
  __syncthreads();
  unsigned wprefix = 0u;
  for (int w = 0; w < wave; ++w) wprefix += woff[w];
  unsigned lprefix = (unsigned)__popcll(bal & ((1ull << lane) - 1ull));

  if (pos) {
    __builtin_prefetch(pred_deltas + (size_t)i * 6, 0, 0);  // global_prefetch
    unsigned rank = blockoffsets[blockIdx.x] + wprefix + lprefix;
    if (rank < NPOS) {
      const float* p = pred_deltas + (size_t)i * 6;
      const float* t = target_deltas + (size_t)rank * 6;
      float s = 0.0f;
#pragma unroll
      for (int k = 0; k < 6; ++k) {
        float d = fabsf(p[k] - t[k]);
        s += (d < 1.0f) ? 0.5f * d * d : d - 0.5f;
      }
      atomicAdd(bbox_sum, s);
    }
  }

  if (i < n && m == -1) {
    float logp0;
    if (keys) {                                  // fast path: 4 B/anchor
      logp0 = key_to_float(keys[i]);
    } else {                                     // fallback: recompute (32 B)
      const float4* row = (const float4*)(logits + (size_t)i * NCLS);
      float4 r0 = row[0], r1 = row[1];
      float l[NCLS] = {r0.x, r0.y, r0.z, r0.w, r1.x, r1.y, r1.z, r1.w};
      float mx = l[0];
#pragma unroll
      for (int k = 1; k < NCLS; ++k) mx = fmaxf(mx, l[k]);
      float s = 0.0f;
#pragma unroll
      for (int k = 0; k < NCLS; ++k) s += __expf(l[k] - mx);
      logp0 = l[0] - (mx + __logf(s));
    }
    unsigned bin = order_key(logp0) >> 20;
    unsigned thr = ctrl[0], quota = ctrl[1];
    bool sel = (bin < thr) ||
               (bin == thr && atomicAdd(&counters[1], 1u) < quota);
    if (sel) {
      unsigned slot = atomicAdd(&counters[0], 1u);
      if (slot < POOLSZ) pool_vals[slot] = -logp0;   // CE of class 0
    }
  }
}

// Pass D (1 block): deterministic 2048-subsample of the pool, WMMA block
// reduce, combine scalars, write the two outputs.
__global__ void __launch_bounds__(BLK) pass_d(const float* __restrict__ pool_vals,
                                              const unsigned* __restrict__ counters,
                                              const float* __restrict__ pos_sum,
                                              const float* __restrict__ bbox_sum,
                                              const int* __restrict__ n_pos_p,
                                              float* __restrict__ out) {
  unsigned pool_n = counters[0];
  if (pool_n > POOLSZ) pool_n = POOLSZ;
  if (pool_n == 0) pool_n = 1;
  float s = 0.0f;
  for (int k = threadIdx.x; k < NPOS; k += BLK) {
    unsigned j = (48271u * (unsigned)k + 7u) % POOLSZ;  // bijection on Z_40960
    if (j >= pool_n) j %= pool_n;
    s += pool_vals[j];
  }
  __shared__ float wsum[BLK / 32];
  float wtot = wave_reduce_wmma(s);
  int wave = threadIdx.x >> 5, lane = threadIdx.x & 31;
  if (lane == 0) wsum[wave] = wtot;
  __syncthreads();
  if (threadIdx.x == 0) {
    float neg_sum = 0.0f;
    for (int w = 0; w < BLK / 32; ++w) neg_sum += wsum[w];
    float n_pos = (float)(*n_pos_p);
    float pos_loss = pos_sum[0] / n_pos;
    float neg_loss = neg_sum / (float)NPOS;
    out[0] = 0.5f * (pos_loss + neg_loss);
    out[1] = bbox_sum[0] / (float)(NPOS * 6);
  }
}

extern "C" void kernel_launch(void* const* d_in, const int* in_sizes, int n_in,
                              void* d_out, int out_size, void* d_ws, size_t ws_size,
                              hipStream_t stream) {
  const int*   matches       = (const int*)d_in[0];
  const float* logits        = (const float*)d_in[1];
  const float* pred_deltas   = (const float*)d_in[2];
  const float* target_deltas = (const float*)d_in[3];
  const int*   n_pos_p       = (const int*)d_in[4];
  const int n    = in_sizes[0];
  const int nblk = (n + BLK - 1) / BLK;

  // Workspace layout (4-byte words):
  //  [0] pos_sum(f32) [1] bbox_sum(f32) [2..3] counters [4..5] ctrl [6..7] pad
  //  [8 .. 8+4095] hist, then blockcounts[nblk], blockoffsets[nblk],
  //  pool_vals[POOLSZ], and (if ws_size allows) keys[n].
  unsigned char* w = (unsigned char*)d_ws;
  float*    pos_sum      = (float*)(w + 0);
  float*    bbox_sum     = (float*)(w + 4);
  unsigned* counters     = (unsigned*)(w + 8);
  unsigned* ctrl         = (unsigned*)(w + 16);
  unsigned* hist         = (unsigned*)(w + 32);
  unsigned* blockcounts  = hist + NBINS;
  unsigned* blockoffsets = blockcounts + nblk;
  float*    pool_vals    = (float*)(blockoffsets + nblk);
  unsigned* keys         = (unsigned*)(pool_vals + POOLSZ);
  size_t need_keys = 32 + (size_t)NBINS * 4 + (size_t)nblk * 8
                   + (size_t)POOLSZ * 4 + (size_t)n * 4;
  if (ws_size < need_keys) keys = nullptr;   // fallback: recompute in pass C
  (void)n_in; (void)out_size;

  const int nzero = 8 + NBINS;  // header + histogram
  zero_ws<<<(nzero + BLK - 1) / BLK, BLK, 0, stream>>>((unsigned*)w, nzero);
  pass_a<<<nblk, BLK, 0, stream>>>(matches, logits, hist, blockcounts,
                                   pos_sum, keys, n);
  pass_b<<<1, BLK, 0, stream>>>(blockcounts, blockoffsets, nblk, hist, ctrl);
  pass_c<<<nblk, BLK, 0, stream>>>(matches, logits, keys, pred_deltas,
                                   target_deltas, blockoffsets, ctrl, counters,
                                   pool_vals, bbox_sum, n);
  pass_d<<<1, BLK, 0, stream>>>(pool_vals, counters, pos_sum, bbox_sum,
                                n_pos_p, (float*)d_out);
}